// Align_Loss_62749472195497
// MI455X (gfx1250) — compile-verified
//
#include <hip/hip_runtime.h>
#include <math.h>

// Problem constants (fixed by setup_inputs in the reference).
#define B_      16
#define LAYERS_ 6
#define L_      200
#define T_      1000
#define V_      5000
#define N_      401     // 2L+1 CTC lattice states
#define NP_     404     // padded lp_states row stride (16B aligned rows)
#define NEG_INF_F (-__builtin_inff())

typedef float v2f __attribute__((ext_vector_type(2)));
typedef float v8f __attribute__((ext_vector_type(8)));

// ---------------------------------------------------------------------------
// Kernel 1: ylens[b] = count(ali_beg[b,:] != -1)
// ---------------------------------------------------------------------------
__global__ void k_init(const int* __restrict__ ali_beg, int* __restrict__ ylens) {
    int b = threadIdx.x;
    if (b < B_) {
        int c = 0;
        for (int l = 0; l < L_; ++l) c += (ali_beg[b * L_ + l] != -1) ? 1 : 0;
        ylens[b] = c;
    }
}

// ---------------------------------------------------------------------------
// Kernel 2: fused log-softmax + label gather.
// One 256-thread block per (b,t) row: row read ONCE into registers,
// logsumexp, then lp_states[b,t,n] = row[label(n)] - lse.
// ---------------------------------------------------------------------------
__global__ __launch_bounds__(256) void k_lse_gather(
        const float* __restrict__ ctc_out,
        const int*   __restrict__ ctc_label,
        float*       __restrict__ lp) {
    const int bt  = blockIdx.x;          // 0 .. B*T-1
    const int b   = bt / T_;
    const int tid = threadIdx.x;
    const float* row = ctc_out + (size_t)bt * V_;

    // 5 chunks of 1024 floats (256 threads * float4) cover 5120 >= 5000.
    float x[20];
    const float4* row4 = (const float4*)row;
#pragma unroll
    for (int w = 0; w < 4; ++w) {
        float4 v = row4[w * 256 + tid];
        x[w * 4 + 0] = v.x; x[w * 4 + 1] = v.y;
        x[w * 4 + 2] = v.z; x[w * 4 + 3] = v.w;
    }
    {
        int base = 4096 + tid * 4;
#pragma unroll
        for (int j = 0; j < 4; ++j)
            x[16 + j] = (base + j < V_) ? row[base + j] : NEG_INF_F;
    }

    float m = NEG_INF_F;
#pragma unroll
    for (int i = 0; i < 20; ++i) m = fmaxf(m, x[i]);

    __shared__ float redm[8], reds[8];
    __shared__ float mSh, sSh;
#pragma unroll
    for (int o = 16; o > 0; o >>= 1) m = fmaxf(m, __shfl_xor(m, o, 32));
    if ((tid & 31) == 0) redm[tid >> 5] = m;
    __syncthreads();
    if (tid == 0) {
        float v = redm[0];
        for (int i = 1; i < 8; ++i) v = fmaxf(v, redm[i]);
        mSh = v;
    }
    __syncthreads();
    const float M = mSh;

    float s = 0.0f;
#pragma unroll
    for (int i = 0; i < 20; ++i) s += __expf(x[i] - M);   // exp(-inf)=0 for pad
#pragma unroll
    for (int o = 16; o > 0; o >>= 1) s += __shfl_xor(s, o, 32);
    if ((tid & 31) == 0) reds[tid >> 5] = s;
    __syncthreads();
    if (tid == 0) {
        float v = reds[0];
        for (int i = 1; i < 8; ++i) v += reds[i];
        sSh = v;
    }
    __syncthreads();
    const float lse = M + __logf(sSh);

    const int* lab = ctc_label + b * L_;
    float* lpr = lp + (size_t)bt * NP_;
    for (int n = tid; n < NP_; n += 256) {
        float v = 0.0f;
        if (n < N_) {
            int li = (n & 1) ? lab[n >> 1] : 0;
            v = row[li] - lse;            // row still hot in L0/L2
        }
        lpr[n] = v;
    }
}

// ---------------------------------------------------------------------------
// Kernel 3: Viterbi forward + backtrace, one workgroup per batch item.
// Thread k owns states n0=2k, n1=2k+1. Ping-pong alpha in LDS; the entire
// backpointer tensor is nibble-packed into ~200 KB of CDNA5's 320 KB LDS,
// so the 999-step backtrace runs at LDS latency instead of L2 latency.
// ---------------------------------------------------------------------------
__global__ __launch_bounds__(256) void k_viterbi(
        const float* __restrict__ lp,
        const int*   __restrict__ ctc_label,
        float*       __restrict__ ali) {
    extern __shared__ unsigned char smem[];
    float* bufA = (float*)smem;                 // 408 floats, states at [2+n]
    float* bufB = bufA + 408;                   // 408 floats
    float* alignLds = bufB + 408;               // 208 floats
    unsigned char* offs = (unsigned char*)(alignLds + 208); // 999 * 204 bytes

    const int b = blockIdx.x;
    const int k = threadIdx.x;
    const float* lpb = lp + (size_t)b * T_ * NP_;

    for (int i = k; i < 408; i += 256) { bufA[i] = NEG_INF_F; bufB[i] = NEG_INF_F; }
    __syncthreads();
    if (k < 2) bufA[2 + k] = lpb[k];            // a0: states 0,1 = lp[t=0]
    __syncthreads();

    // two_way for odd state n1=2k+1: n1==1 or labels repeat
    bool twoway1 = true;
    if (k >= 1 && k <= 199)
        twoway1 = (ctc_label[b * L_ + k] == ctc_label[b * L_ + k - 1]);

    float* A  = bufA;
    float* Bf = bufB;

    float2 lv = make_float2(0.f, 0.f);
    if (k <= 200) lv = ((const float2*)(lpb + (size_t)NP_))[k];

    for (int t = 1; t < T_; ++t) {
        float2 lnext = make_float2(0.f, 0.f);
        if (t + 1 < T_ && k <= 200) {
            lnext = ((const float2*)(lpb + (size_t)(t + 1) * NP_))[k];
            __builtin_prefetch(lpb + (size_t)(t + 1) * NP_ + 2 * k, 0, 1);
        }
        const float tf = (float)t;

        float pm2 = NEG_INF_F, pm1 = NEG_INF_F, p00 = NEG_INF_F, p01 = NEG_INF_F;
        if (k <= 200) {
            float2 u = ((const float2*)A)[k];       // a[n0-2], a[n0-1]
            float2 w = ((const float2*)A)[k + 1];   // a[n0],   a[n1]
            pm2 = u.x; pm1 = u.y; p00 = w.x; p01 = w.y;
        }

        // even state n0 = 2k (always two-way; n0==0 forced to self)
        int off0; float sel0;
        if (k == 0) { off0 = 0; sel0 = p00; }
        else        { off0 = (p00 > pm1) ? 0 : 1; sel0 = off0 ? pm1 : p00; }

        // odd state n1 = 2k+1
        int off1; float sel1;
        if (twoway1) {
            off1 = (p01 > p00) ? 0 : 1; sel1 = off1 ? p00 : p01;
        } else {
            bool b0 = (p01 > p00) && (p01 > pm1);
            bool b1 = (p00 > p01) && (p00 > pm1);
            off1 = b0 ? 0 : (b1 ? 1 : 2);
            sel1 = b0 ? p01 : (b1 ? p00 : pm1);
        }

        // feasibility: !(n/2 > t || L - n/2 > T - t)
        const float n0h = (float)k;
        const float n1h = (float)k + 0.5f;
        bool f0 = !((n0h > tf) || (200.0f - n0h > 1000.0f - tf));
        bool f1 = !((n1h > tf) || (200.0f - n1h > 1000.0f - tf));
        float a0n = f0 ? (sel0 + lv.x) : NEG_INF_F;
        float a1n = f1 ? (sel1 + lv.y) : NEG_INF_F;

        if (k <= 200) {
            float2 st;
            st.x = a0n;
            st.y = (k <= 199) ? a1n : NEG_INF_F;
            ((float2*)Bf)[k + 1] = st;                  // B[2+n0], B[2+n1]
            int o1s = (k <= 199) ? off1 : 0;
            offs[(size_t)(t - 1) * 204 + k] = (unsigned char)(off0 | (o1s << 4));
        }
        __syncthreads();
        float* tmp = A; A = Bf; Bf = tmp;
        lv = lnext;
    }

    if (k < L_) alignLds[k] = 0.0f;
    __syncthreads();

    if (k == 0) {
        bool use_last = A[2 + 400] > A[2 + 399];
        int pre = use_last ? 400 : 399;
        if (!use_last) alignLds[L_ - 1] = (float)T_;
        for (int t1 = T_ - 1; t1 >= 1; --t1) {
            unsigned char byte = offs[(size_t)(t1 - 1) * 204 + (pre >> 1)];
            int off = (pre & 1) ? ((byte >> 4) & 0xF) : (byte & 0xF);
            pre -= off;
            if (pre & 1) alignLds[pre >> 1] = (float)t1;  // last (smallest t1) wins
        }
    }
    __syncthreads();
    for (int l = k; l < L_; l += 256) ali[b * L_ + l] = alignLds[l];
}

// ---------------------------------------------------------------------------
// Kernel 4: ali_pos = einsum('bloi,i->blo') via V_WMMA_F32_16X16X4_F32,
// then masked (ali_pos - ali)^2 partial sums. One wave per 16-row tile;
// B matrix carries the position vector in column 0 only (matvec through
// the matrix core — memory-bound, so the idle columns are free).
// ---------------------------------------------------------------------------
__global__ __launch_bounds__(32) void k_alipos(
        const float* __restrict__ ali_out,
        const float* __restrict__ ali,
        const int*   __restrict__ ylens,
        float*       __restrict__ partials) {
    const int r0   = blockIdx.x * 16;
    const int lane = threadIdx.x;
    const int half = lane >> 4;      // 0: K=0,1   1: K=2,3
    const int lm   = lane & 15;      // M (A) / N (B) index
    const float* rowp = ali_out + (size_t)(r0 + lm) * T_;

    v8f c = {0.f, 0.f, 0.f, 0.f, 0.f, 0.f, 0.f, 0.f};
    for (int kk = 0; kk < T_; kk += 4) {
        const int i0 = kk + 2 * half;
        v2f a;  a.x = rowp[i0]; a.y = rowp[i0 + 1];          // A: 16x4 tile
        v2f bv;                                               // B: pos in col 0
        bv.x = (lm == 0) ? (float)(i0 + 1) : 0.0f;
        bv.y = (lm == 0) ? (float)(i0 + 2) : 0.0f;
        c = __builtin_amdgcn_wmma_f32_16x16x4_f32(
                false, a, false, bv, (short)0, c, false, false);
    }

    // Column 0 of D lives in lanes 0 (M=j) and 16 (M=8+j), VGPRs j=0..7.
    float local = 0.0f;
    if (lm == 0) {
#pragma unroll
        for (int j = 0; j < 8; ++j) {
            int r  = r0 + half * 8 + j;
            int o  = r % (L_ + 1);
            int bl = r / (L_ + 1);
            int b  = bl / LAYERS_;
            if (o < L_) {                        // drop last output step
                float d = c[j] - ali[b * L_ + o];
                if (o < ylens[b]) local += d * d;   // pad mask
            }
        }
    }
    local += __shfl_xor(local, 16, 32);
    if (lane == 0) partials[blockIdx.x] = local;
}

// ---------------------------------------------------------------------------
// Kernel 5: deterministic final reduction + normalization.
// ---------------------------------------------------------------------------
__global__ __launch_bounds__(256) void k_final(
        const float* __restrict__ partials, int nparts,
        const int*   __restrict__ ylens,
        float*       __restrict__ out) {
    __shared__ float red[256];
    int tid = threadIdx.x;
    float s = 0.0f;
    for (int i = tid; i < nparts; i += 256) s += partials[i];
    red[tid] = s;
    __syncthreads();
    for (int st = 128; st >= 1; st >>= 1) {
        if (tid < st) red[tid] += red[tid + st];
        __syncthreads();
    }
    if (tid == 0) {
        int cnt = 0;
        for (int b = 0; b < B_; ++b) cnt += ylens[b];
        float total = (float)cnt * (float)LAYERS_;
        out[0] = red[0] / total / (float)T_;
    }
}

// ---------------------------------------------------------------------------
extern "C" void kernel_launch(void* const* d_in, const int* in_sizes, int n_in,
                              void* d_out, int out_size, void* d_ws, size_t ws_size,
                              hipStream_t stream) {
    (void)in_sizes; (void)n_in; (void)out_size; (void)ws_size;
    const float* ali_out   = (const float*)d_in[0];
    const int*   ali_beg   = (const int*)  d_in[1];
    // d_in[2]=ali_end, d_in[3]=enc_mask: unused by the loss value
    const float* ctc_out   = (const float*)d_in[4];
    const int*   ctc_label = (const int*)  d_in[5];
    float* out = (float*)d_out;

    unsigned char* ws = (unsigned char*)d_ws;
    size_t off = 0;
    float* lp = (float*)(ws + off);           off += (size_t)B_ * T_ * NP_ * sizeof(float);
    float* ali = (float*)(ws + off);          off += (size_t)B_ * L_ * sizeof(float);
    int* ylens = (int*)(ws + off);            off += 64;
    const int nparts = (B_ * LAYERS_ * (L_ + 1)) / 16;   // 1206 tiles
    float* partials = (float*)(ws + off);     off += (size_t)nparts * sizeof(float);

    k_init<<<1, 32, 0, stream>>>(ali_beg, ylens);
    k_lse_gather<<<B_ * T_, 256, 0, stream>>>(ctc_out, ctc_label, lp);

    const size_t lds_bytes = (408 + 408 + 208) * sizeof(float) + (size_t)999 * 204;
    k_viterbi<<<B_, 256, lds_bytes, stream>>>(lp, ctc_label, ali);

    k_alipos<<<nparts, 32, 0, stream>>>(ali_out, ali, ylens, partials);
    k_final<<<1, 256, 0, stream>>>(partials, nparts, ylens, out);
}